// SparseTopKAttentionGATLayer_69441031242186
// MI455X (gfx1250) — compile-verified
//
#include <hip/hip_runtime.h>
#include <hip/hip_bf16.h>
#include <math.h>

// ---------------------------------------------------------------------------
// SparseTopKAttentionGATLayer for MI455X (gfx1250)
//   x:(N,128) f32, W:(128,128) f32, att:(4,64) f32, edge_index:(2,E) i32
//   out:(N,128) f32
// ---------------------------------------------------------------------------

typedef __attribute__((ext_vector_type(2))) float v2f;
typedef __attribute__((ext_vector_type(8))) float v8f;

#define IN_CH   128
#define HEADS   4
#define OUT_CH  32
#define TOPK    8

// ---------------------------------------------------------------------------
// Kernel 1: h = x @ W via v_wmma_f32_16x16x4_f32 (exact fp32 accumulation).
// Block = 256 threads = 8 waves; each wave owns one 16x16 output tile
// (wave id selects the column tile). The 16x128 stripe of x is staged in LDS
// (rows padded 128->132 floats; A-fragment reads pair into ds_load_2addr_b64
// with no bank conflicts). W is read directly from global: 64KB, L2/L0
// resident, shared by all blocks.
// ---------------------------------------------------------------------------
__global__ __launch_bounds__(256) void gat_gemm_wmma(
    const float* __restrict__ x, const float* __restrict__ W,
    float* __restrict__ h, int n_nodes)
{
    __shared__ float xs[16 * 132];
    const int rowBase = blockIdx.x * 16;
    const int t = threadIdx.x;
    const bool fullTile = (rowBase + 16 <= n_nodes);

    // Stage 16x128 stripe of x (zero-padded past n_nodes).
    if (fullTile) {
#pragma unroll
        for (int i = 0; i < 8; ++i) {
            int idx = i * 256 + t;
            int r = idx >> 7;
            int c = idx & 127;
            xs[r * 132 + c] = x[(size_t)(rowBase + r) * IN_CH + c];
        }
    } else {
#pragma unroll
        for (int i = 0; i < 8; ++i) {
            int idx = i * 256 + t;
            int r = idx >> 7;
            int c = idx & 127;
            int row = rowBase + r;
            xs[r * 132 + c] = (row < n_nodes) ? x[(size_t)row * IN_CH + c] : 0.0f;
        }
    }
    __syncthreads();

    const int wave    = t >> 5;          // 0..7 -> column tile
    const int lane    = t & 31;
    const int half    = lane >> 4;       // 0: K={0,1}; 1: K={2,3} (A frag)
    const int l16     = lane & 15;
    const int colBase = wave * 16;

    v8f acc = {};
#pragma unroll 4
    for (int k0 = 0; k0 < IN_CH; k0 += 4) {
        v2f a, b;
        // A 16x4: lanes 0-15 hold M=l16, K={k0,k0+1}; lanes 16-31 K={k0+2,k0+3}
        a.x = xs[l16 * 132 + k0 + 2 * half + 0];
        a.y = xs[l16 * 132 + k0 + 2 * half + 1];
        // B 4x16 row-striped across lanes: VGPR0 -> K=k0 (lo half) / k0+2 (hi),
        //                                  VGPR1 -> K=k0+1 / k0+3
        b.x = W[(size_t)(k0 + 2 * half + 0) * IN_CH + colBase + l16];
        b.y = W[(size_t)(k0 + 2 * half + 1) * IN_CH + colBase + l16];
        acc = __builtin_amdgcn_wmma_f32_16x16x4_f32(
            /*neg_a=*/false, a, /*neg_b=*/false, b,
            /*c_mod=*/(short)0, acc, /*reuse_a=*/false, /*reuse_b=*/false);
    }

    // D 16x16: VGPR v -> row M = v + 8*half, col = colBase + l16.
    // Fast path: whole tile in-bounds -> no per-store exec-mask juggling.
    float* hp = h + (size_t)(rowBase + 8 * half) * IN_CH + colBase + l16;
    if (fullTile) {
#pragma unroll
        for (int v = 0; v < 8; ++v)
            hp[(size_t)v * IN_CH] = acc[v];
    } else {
#pragma unroll
        for (int v = 0; v < 8; ++v) {
            int row = rowBase + v + 8 * half;
            if (row < n_nodes)
                h[(size_t)row * IN_CH + colBase + l16] = acc[v];
        }
    }
}

// ---------------------------------------------------------------------------
// Kernel 2: per-node attention score vectors.
//   si[n*4+hh] = <h[n,hh,:], att[hh, 0:32]>, sj = <h[n,hh,:], att[hh,32:64]>
// ---------------------------------------------------------------------------
__global__ __launch_bounds__(256) void gat_sisj(
    const float* __restrict__ h, const float* __restrict__ att,
    float* __restrict__ si, float* __restrict__ sj, int n_nodes)
{
    int tid = blockIdx.x * blockDim.x + threadIdx.x;
    if (tid >= n_nodes * HEADS) return;
    int node = tid >> 2;
    int hh   = tid & 3;
    const float* hp = h + (size_t)node * IN_CH + hh * OUT_CH;
    const float* ai = att + hh * (2 * OUT_CH);
    const float* aj = ai + OUT_CH;
    float s1 = 0.0f, s2 = 0.0f;
#pragma unroll
    for (int c = 0; c < OUT_CH; ++c) {
        float v = hp[c];
        s1 += v * ai[c];
        s2 += v * aj[c];
    }
    si[tid] = s1;
    sj[tid] = s2;
}

// ---------------------------------------------------------------------------
// CSR build: zero, count, scan (2-level), scatter.
// Edge e in [0,E): row=ei[e], col=ei[E+e]. e in [E, E+N): self loop (e-E,e-E).
// ---------------------------------------------------------------------------
__global__ void gat_zero_i32(int* __restrict__ p, int n)
{
    int i = blockIdx.x * blockDim.x + threadIdx.x;
    if (i < n) p[i] = 0;
}

__global__ void gat_count(const int* __restrict__ ei, int E, int n_nodes,
                          int* __restrict__ deg)
{
    int e = blockIdx.x * blockDim.x + threadIdx.x;
    int Etot = E + n_nodes;
    if (e >= Etot) return;
    int r = (e < E) ? ei[e] : (e - E);
    atomicAdd(&deg[r], 1);
}

__global__ __launch_bounds__(1024) void gat_scan_block(
    const int* __restrict__ deg, int* __restrict__ offs,
    int* __restrict__ blockSums, int n)
{
    __shared__ int tmp[1024];
    int gid = blockIdx.x * 1024 + threadIdx.x;
    int v = (gid < n) ? deg[gid] : 0;
    tmp[threadIdx.x] = v;
    __syncthreads();
    // Hillis-Steele inclusive scan
    for (int off = 1; off < 1024; off <<= 1) {
        int t = (threadIdx.x >= off) ? tmp[threadIdx.x - off] : 0;
        __syncthreads();
        tmp[threadIdx.x] += t;
        __syncthreads();
    }
    if (gid < n) offs[gid] = tmp[threadIdx.x] - v;   // exclusive
    if (threadIdx.x == 1023) blockSums[blockIdx.x] = tmp[1023];
}

__global__ void gat_scan_sums(int* __restrict__ blockSums, int nb,
                              int* __restrict__ offs, int n, int Etot)
{
    if (blockIdx.x == 0 && threadIdx.x == 0) {
        int acc = 0;
        for (int i = 0; i < nb; ++i) {
            int t = blockSums[i];
            blockSums[i] = acc;
            acc += t;
        }
        offs[n] = Etot;
    }
}

__global__ __launch_bounds__(1024) void gat_scan_add(
    int* __restrict__ offs, const int* __restrict__ blockSums, int n)
{
    int gid = blockIdx.x * 1024 + threadIdx.x;
    if (gid < n) offs[gid] += blockSums[blockIdx.x];
}

__global__ void gat_scatter(const int* __restrict__ ei, int E, int n_nodes,
                            const int* __restrict__ offs,
                            int* __restrict__ cursor, int* __restrict__ csr_col)
{
    int e = blockIdx.x * blockDim.x + threadIdx.x;
    int Etot = E + n_nodes;
    if (e >= Etot) return;
    int r, c;
    if (e < E) { r = ei[e]; c = ei[E + e]; }
    else       { r = e - E; c = r; }
    int pos = atomicAdd(&cursor[r], 1);
    csr_col[offs[r] + pos] = c;
}

// ---------------------------------------------------------------------------
// Kernel 7: per-(node,head) top-8 + softmax + aggregation + ELU.
// Streaming top-8 via 8-register insertion list; scores recomputed from
// si/sj (cheaper than storing E*H floats). Ties have zero probability with
// continuous random scores, so "s >= thr" matches the reference mask.
// ---------------------------------------------------------------------------
__global__ __launch_bounds__(128) void gat_attn_agg(
    const float* __restrict__ h, const float* __restrict__ si,
    const float* __restrict__ sj, const int* __restrict__ offs,
    const int* __restrict__ csr_col, float* __restrict__ out, int n_nodes)
{
    int tid = blockIdx.x * blockDim.x + threadIdx.x;
    if (tid >= n_nodes * HEADS) return;
    int node = tid >> 2;
    int hh   = tid & 3;
    int off  = offs[node];
    int deg  = offs[node + 1] - off;
    float s_i = si[tid];

    float top[TOPK];
#pragma unroll
    for (int j = 0; j < TOPK; ++j) top[j] = -INFINITY;

    // Pass 1: top-8 threshold + max (prefetch h rows for pass 2).
    for (int i = 0; i < deg; ++i) {
        int c = csr_col[off + i];
        float s = s_i + sj[c * HEADS + hh];
        s = fmaxf(s, 0.2f * s);                 // leaky_relu(0.2)
        if (s > top[TOPK - 1]) {
            top[TOPK - 1] = s;
#pragma unroll
            for (int j = TOPK - 1; j > 0; --j)
                if (top[j] > top[j - 1]) {
                    float t = top[j - 1]; top[j - 1] = top[j]; top[j] = t;
                }
        }
        __builtin_prefetch(&h[(size_t)c * IN_CH + hh * OUT_CH], 0, 1);
    }
    float m   = top[0];
    float thr = (deg <= TOPK) ? -INFINITY : top[TOPK - 1];

    // Pass 2: fused exp-weight + aggregation (out = (sum ez*h) / sum ez).
    float denom = 0.0f;
    float acc[OUT_CH];
#pragma unroll
    for (int j = 0; j < OUT_CH; ++j) acc[j] = 0.0f;

    for (int i = 0; i < deg; ++i) {
        int c = csr_col[off + i];
        float s = s_i + sj[c * HEADS + hh];
        s = fmaxf(s, 0.2f * s);
        if (s >= thr) {
            float ez = expf(s - m);
            denom += ez;
            const float4* hp = reinterpret_cast<const float4*>(
                h + (size_t)c * IN_CH + hh * OUT_CH);
#pragma unroll
            for (int q = 0; q < 8; ++q) {
                float4 v = hp[q];
                acc[q * 4 + 0] += ez * v.x;
                acc[q * 4 + 1] += ez * v.y;
                acc[q * 4 + 2] += ez * v.z;
                acc[q * 4 + 3] += ez * v.w;
            }
        }
    }

    float inv = 1.0f / denom;
    float* op = out + (size_t)node * IN_CH + hh * OUT_CH;
#pragma unroll
    for (int j = 0; j < OUT_CH; ++j) {
        float o = acc[j] * inv;
        op[j] = (o > 0.0f) ? o : (expf(o) - 1.0f);   // elu(alpha=1)
    }
}

// ---------------------------------------------------------------------------
// Host-side launch (graph-capture safe: all launches on `stream`, no sync,
// no allocation; scratch carved out of d_ws; mutable scratch re-zeroed each
// call so replays are deterministic).
// ---------------------------------------------------------------------------
static inline size_t align256(size_t v) { return (v + 255) & ~(size_t)255; }

extern "C" void kernel_launch(void* const* d_in, const int* in_sizes, int n_in,
                              void* d_out, int out_size, void* d_ws, size_t ws_size,
                              hipStream_t stream) {
    const float* x   = (const float*)d_in[0];   // (N,128)
    const float* W   = (const float*)d_in[1];   // (128,128)
    const float* att = (const float*)d_in[2];   // (4,64)
    const int*   ei  = (const int*)  d_in[3];   // (2,E)

    const int N    = in_sizes[0] / IN_CH;
    const int E    = in_sizes[3] / 2;
    const int Etot = E + N;

    // Workspace carve-out (~31 MB total).
    char*  p        = (char*)d_ws;
    float* h        = (float*)p;  p += align256((size_t)N * IN_CH * sizeof(float));
    float* si       = (float*)p;  p += align256((size_t)N * HEADS * sizeof(float));
    float* sj       = (float*)p;  p += align256((size_t)N * HEADS * sizeof(float));
    int*   deg      = (int*)p;    p += align256((size_t)N * sizeof(int));
    int*   cursor   = (int*)p;    p += align256((size_t)N * sizeof(int));
    int*   offs     = (int*)p;    p += align256((size_t)(N + 1) * sizeof(int));
    int*   bsums    = (int*)p;    p += align256(1024 * sizeof(int));
    int*   csr_col  = (int*)p;    p += align256((size_t)Etot * sizeof(int));
    (void)ws_size; (void)n_in; (void)out_size;

    // 1. h = x @ W (WMMA)
    gat_gemm_wmma<<<(N + 15) / 16, 256, 0, stream>>>(x, W, h, N);

    // 2. per-node score vectors
    gat_sisj<<<(N * HEADS + 255) / 256, 256, 0, stream>>>(h, att, si, sj, N);

    // 3. CSR build
    gat_zero_i32<<<(N + 255) / 256, 256, 0, stream>>>(deg, N);
    gat_zero_i32<<<(N + 255) / 256, 256, 0, stream>>>(cursor, N);
    gat_count<<<(Etot + 255) / 256, 256, 0, stream>>>(ei, E, N, deg);

    const int nb = (N + 1023) / 1024;
    gat_scan_block<<<nb, 1024, 0, stream>>>(deg, offs, bsums, N);
    gat_scan_sums<<<1, 1, 0, stream>>>(bsums, nb, offs, N, Etot);
    gat_scan_add<<<nb, 1024, 0, stream>>>(offs, bsums, N);

    gat_scatter<<<(Etot + 255) / 256, 256, 0, stream>>>(ei, E, N, offs, cursor, csr_col);

    // 4. top-8 softmax aggregation + ELU
    gat_attn_agg<<<(N * HEADS + 127) / 128, 128, 0, stream>>>(
        h, si, sj, offs, csr_col, (float*)d_out, N);
}